// StationsLoss_spl_25563645345793
// MI455X (gfx1250) — compile-verified
//
#include <hip/hip_runtime.h>
#include <stdint.h>

#define TPB    256u
#define TILE   8192u
#define CHUNKS (TILE / TPB)   // 32
#define RADIX  256u

typedef unsigned int  u32;
typedef unsigned char u8;

typedef __attribute__((ext_vector_type(2))) float v2f;
typedef __attribute__((ext_vector_type(8))) float v8f;

__device__ __forceinline__ float loss_fn(float p, float t) {
  float s = 2.0f * fabsf(t - p) / (fabsf(t) + fabsf(p));
  bool valid = (t == t) && (t != 0.0f);   // !isnan(tgt) && tgt != 0
  return valid ? s : 3.0f;
}

__device__ __forceinline__ u32 key_of(float f) {
  u32 u = __float_as_uint(f);
  return u ^ ((u32)((int)u >> 31) | 0x80000000u);  // order-preserving float->uint
}

// ---------------------------------------------------------------- build keys
__global__ __launch_bounds__(TPB) void buildKeysKernel(
    const float* __restrict__ pre, const float* __restrict__ tgt,
    u32* __restrict__ keys, u32* __restrict__ vals, u32 N) {
  u32 i0 = (blockIdx.x * TPB + threadIdx.x) * 4u;
#pragma unroll
  for (u32 k = 0; k < 4; k++) {
    u32 i = i0 + k;
    if (i < N) {
      keys[i] = key_of(loss_fn(pre[i], tgt[i]));
      vals[i] = i;
    }
  }
}

// --------------------------------------------- per-tile histogram (async LDS)
__global__ __launch_bounds__(TPB) void histKernel(
    const u32* __restrict__ keys, u32* __restrict__ histMat,
    u32 shift, u32 numTiles, u32 N) {
  __shared__ u32 h[RADIX];
  __shared__ u32 stage[2048];  // 8KB staging, 4 chunks per tile
  u32 t = threadIdx.x;
  h[t] = 0;
  __syncthreads();
  u32 base = blockIdx.x * TILE;
  if (base + TILE <= N) {
    u32 sbase = (u32)(uintptr_t)(&stage[0]);  // LDS byte address = low 32 bits
    u32 l0 = sbase + (t * 8u) * 4u;
    for (u32 c = 0; c < 4; c++) {
      const u32* gp0 = keys + base + c * 2048u + t * 8u;
      // make sure prior ds reads of stage finished before DMA overwrites it
      asm volatile("s_wait_dscnt 0" ::: "memory");
      asm volatile("global_load_async_to_lds_b128 %0, %1, off"
                   :: "v"(l0), "v"(gp0) : "memory");
      asm volatile("global_load_async_to_lds_b128 %0, %1, off"
                   :: "v"(l0 + 16u), "v"(gp0 + 4) : "memory");
      asm volatile("s_wait_asynccnt 0" ::: "memory");
#pragma unroll
      for (u32 k = 0; k < 8; k++) {
        u32 d = (stage[t * 8u + k] >> shift) & 255u;
        atomicAdd(&h[d], 1u);
      }
    }
  } else {
    for (u32 c = 0; c < CHUNKS; c++) {
      u32 i = base + c * TPB + t;
      if (i < N) atomicAdd(&h[(keys[i] >> shift) & 255u], 1u);
    }
  }
  __syncthreads();
  histMat[t * numTiles + blockIdx.x] = h[t];
}

// ------------------------------- generic in-place exclusive row scan + totals
__global__ __launch_bounds__(TPB) void rowScanKernel(
    u32* __restrict__ mat, u32 rowLen, u32* __restrict__ rowTot) {
  __shared__ u32 s[TPB];
  u32 row = blockIdx.x, t = threadIdx.x;
  u32 base = row * rowLen;
  u32 running = 0;
  for (u32 c0 = 0; c0 < rowLen; c0 += TPB) {
    u32 idx = c0 + t;
    u32 v = (idx < rowLen) ? mat[base + idx] : 0u;
    s[t] = v;
    __syncthreads();
    for (u32 off = 1; off < TPB; off <<= 1) {
      u32 x = (t >= off) ? s[t - off] : 0u;
      __syncthreads();
      s[t] += x;
      __syncthreads();
    }
    u32 incl = s[t];
    u32 tot = s[TPB - 1];
    if (idx < rowLen) mat[base + idx] = running + incl - v;
    running += tot;
    __syncthreads();
  }
  if (t == 0) rowTot[row] = running;
}

// ------------- 256-entry exclusive scan of digit totals via v_wmma (exact f32)
// excl = X*U - X + L*rowTotals ; all values <= 2^24 so fp32 arithmetic is exact.
__global__ __launch_bounds__(32) void scanDigitsWMMA(
    const u32* __restrict__ tot, u32* __restrict__ off) {
  __shared__ float rowTot[16];
  u32 lane = threadIdx.x;
  int n  = (int)(lane & 15u);          // column / M index (A layout: M = lane&15)
  int kb = (int)(((lane >> 4) & 1u) << 1);  // lanes 16-31 hold K = 2,3

  v8f acc = {};
#pragma unroll
  for (int j = 0; j < 4; j++) {        // inc = X * U  (U[k][c] = k <= c)
    v2f a, b;
    a.x = (float)tot[n * 16 + 4 * j + kb + 0];
    a.y = (float)tot[n * 16 + 4 * j + kb + 1];
    b.x = ((4 * j + kb + 0) <= n) ? 1.0f : 0.0f;
    b.y = ((4 * j + kb + 1) <= n) ? 1.0f : 0.0f;
    acc = __builtin_amdgcn_wmma_f32_16x16x4_f32(false, a, false, b,
                                                (short)0, acc, false, false);
  }
  if ((lane & 15u) == 15u) {           // column 15 holds inclusive row sums
    u32 mbase = (lane >> 4) * 8u;
#pragma unroll
    for (int v = 0; v < 8; v++) rowTot[mbase + v] = acc[v];
  }
  __syncthreads();
  v8f acc2 = {};
#pragma unroll
  for (int j = 0; j < 4; j++) {        // rowOffset = L * T  (L[r][r'] = r' < r)
    v2f a, b;
    int k0 = 4 * j + kb;
    a.x = (k0 + 0 < n) ? 1.0f : 0.0f;
    a.y = (k0 + 1 < n) ? 1.0f : 0.0f;
    b.x = rowTot[k0 + 0];
    b.y = rowTot[k0 + 1];
    acc2 = __builtin_amdgcn_wmma_f32_16x16x4_f32(false, a, false, b,
                                                 (short)0, acc2, false, false);
  }
  u32 mofs = (lane >> 4) * 8u;         // C/D layout: M = v + 8*(lane>=16), N = lane&15
#pragma unroll
  for (int v = 0; v < 8; v++) {
    u32 M = (u32)v + mofs;
    float x = (float)tot[M * 16 + (u32)n];
    float excl = acc[v] - x + acc2[v];
    off[M * 16 + (u32)n] = (u32)(excl + 0.5f);
  }
}

// ----------------------------------- stable scatter via wave32 ballot multisplit
template <bool LAST>
__global__ __launch_bounds__(TPB) void scatterKernel(
    const u32* __restrict__ keyIn, const u32* __restrict__ valIn,
    u32* __restrict__ keyOut, u32* __restrict__ valOut, u32* __restrict__ inv,
    const u32* __restrict__ histScanned, const u32* __restrict__ digitOffset,
    u32 shift, u32 numTiles, u32 N) {
  __shared__ u32 gBase[RADIX];
  __shared__ u32 running[RADIX];
  __shared__ u32 wHist[8 * RADIX];
  u32 t = threadIdx.x, tile = blockIdx.x;
  gBase[t] = digitOffset[t] + histScanned[t * numTiles + tile];
  running[t] = 0;
  __syncthreads();
  u32 lane = t & 31u, wave = t >> 5;
  u32 lt = (1u << lane) - 1u;
  u32 base = tile * TILE;
  for (u32 c = 0; c < CHUNKS; c++) {
    u32 i = base + c * TPB + t;
    bool inb = (i < N);
    u32 key = 0u, val = 0u;
    if (inb) { key = keyIn[i]; val = valIn[i]; }
    u32 d = (key >> shift) & 255u;
    u32 m = __builtin_amdgcn_ballot_w32(inb);   // peer mask, active lanes only
#pragma unroll
    for (u32 b = 0; b < 8; b++) {
      u32 bal = __builtin_amdgcn_ballot_w32((bool)((d >> b) & 1u));
      m &= ((d >> b) & 1u) ? bal : ~bal;
    }
    u32 rIn = (u32)__popc(m & lt);
#pragma unroll
    for (u32 k = 0; k < 8; k++) wHist[k * RADIX + t] = 0;
    __syncthreads();
    if (inb && rIn == 0) wHist[wave * RADIX + d] = (u32)__popc(m);
    __syncthreads();
    {   // thread t scans digit t across the 8 waves (stable across waves)
      u32 s = running[t];
#pragma unroll
      for (u32 w8 = 0; w8 < 8; w8++) {
        u32 cnt = wHist[w8 * RADIX + t];
        wHist[w8 * RADIX + t] = s;
        s += cnt;
      }
      running[t] = s;
    }
    __syncthreads();
    if (inb) {
      u32 dest = gBase[d] + wHist[wave * RADIX + d] + rIn;
      keyOut[dest] = key;
      valOut[dest] = val;
      if (LAST) inv[val] = dest;   // inverse permutation
    }
    __syncthreads();
  }
}

// -------------------- e[w] = level[inv[w]] (byte) + per-tile 5-channel sums
__global__ __launch_bounds__(TPB) void tileSumKernel(
    const u32* __restrict__ inv, const int* __restrict__ level,
    u8* __restrict__ e, u32* __restrict__ tileSum, u32 numTiles, u32 N) {
  __shared__ u32 s5[5];
  u32 t = threadIdx.x;
  if (t < 5) s5[t] = 0;
  __syncthreads();
  u32 base = blockIdx.x * TILE;
  u32 c0 = 0, c1 = 0, c2 = 0, c3 = 0, c4 = 0;
  for (u32 c = 0; c < CHUNKS; c++) {
    u32 wdx = base + c * TPB + t;
    if (wdx < N) {
      int lv = level[inv[wdx]];
      e[wdx] = (u8)lv;
      c0 += (lv == 0); c1 += (lv == 1); c2 += (lv == 2);
      c3 += (lv == 3); c4 += (lv == 4);
    }
  }
  atomicAdd(&s5[0], c0); atomicAdd(&s5[1], c1); atomicAdd(&s5[2], c2);
  atomicAdd(&s5[3], c3); atomicAdd(&s5[4], c4);
  __syncthreads();
  if (t < 5) tileSum[t * numTiles + blockIdx.x] = s5[t];
}

// --------------------------- write the 5 exclusive per-level prefix arrays
__global__ __launch_bounds__(TPB) void writePKernel(
    const u8* __restrict__ e, const u32* __restrict__ tileSumScanned,
    u32* __restrict__ P0, u32* __restrict__ P1, u32* __restrict__ P2,
    u32* __restrict__ P3, u32* __restrict__ P4, u32 numTiles, u32 N) {
  __shared__ u32 sc[5][TPB + 1];
  u32 t = threadIdx.x, tile = blockIdx.x;
  u32 base = tile * TILE + t * 32u;   // contiguous per-thread ownership (stable order)
  u32 c0 = 0, c1 = 0, c2 = 0, c3 = 0, c4 = 0;
  u8 loc[32];
#pragma unroll
  for (u32 k = 0; k < 32; k++) {
    u32 wdx = base + k;
    u8 lv = (wdx < N) ? e[wdx] : (u8)255;
    loc[k] = lv;
    c0 += (lv == 0); c1 += (lv == 1); c2 += (lv == 2);
    c3 += (lv == 3); c4 += (lv == 4);
  }
  sc[0][t] = c0; sc[1][t] = c1; sc[2][t] = c2; sc[3][t] = c3; sc[4][t] = c4;
  __syncthreads();
  if (t < 5) {   // serial exclusive scan across 256 threads, one channel each
    u32 run = tileSumScanned[t * numTiles + tile];
    for (u32 i = 0; i < TPB; i++) { u32 v = sc[t][i]; sc[t][i] = run; run += v; }
  }
  __syncthreads();
  u32 r0 = sc[0][t], r1 = sc[1][t], r2 = sc[2][t], r3 = sc[3][t], r4 = sc[4][t];
#pragma unroll
  for (u32 k = 0; k < 32; k++) {
    u32 wdx = base + k;
    if (wdx < N) {
      P0[wdx] = r0; P1[wdx] = r1; P2[wdx] = r2; P3[wdx] = r3; P4[wdx] = r4;
      u8 lv = loc[k];
      r0 += (lv == 0); r1 += (lv == 1); r2 += (lv == 2);
      r3 += (lv == 3); r4 += (lv == 4);
    }
  }
}

// ------------------------------------------------------------------- finalize
__global__ __launch_bounds__(TPB) void finalKernel(
    const float* __restrict__ pre, const float* __restrict__ tgt,
    const int* __restrict__ level, const u32* __restrict__ idx,
    const u32* __restrict__ P0, const u32* __restrict__ P1,
    const u32* __restrict__ P2, const u32* __restrict__ P3,
    const u32* __restrict__ P4, float* __restrict__ out, u32 N) {
  u32 i = blockIdx.x * TPB + threadIdx.x;
  if (i >= N) return;
  float l = loss_fn(pre[i], tgt[i]);
  int lv = level[i];
  u32 v = idx[i];
  const u32* P = (lv == 0) ? P0 : (lv == 1) ? P1 : (lv == 2) ? P2 : (lv == 3) ? P3 : P4;
  float rf = (float)P[v];
  float th = 0.5f + 1.0f / (sqrtf(rf + 1.0f) + sqrtf(rf));
  out[i] = (l < th) ? 0.0f : 1.0f;   // 1 - v
}

extern "C" void kernel_launch(void* const* d_in, const int* in_sizes, int n_in,
                              void* d_out, int out_size, void* d_ws, size_t ws_size,
                              hipStream_t stream) {
  const float* pre = (const float*)d_in[0];
  const float* tgt = (const float*)d_in[1];
  const int* level = (const int*)d_in[2];
  u32 N = (u32)in_sizes[0];
  u32 numTiles = (N + TILE - 1) / TILE;

  char* w = (char*)d_ws;
  auto alloc = [&](size_t bytes) -> char* {
    char* p = w; w += (bytes + 255) & ~(size_t)255; return p;
  };
  u32* KA  = (u32*)alloc((size_t)N * 4);
  u32* KB  = (u32*)alloc((size_t)N * 4);
  u32* VA  = (u32*)alloc((size_t)N * 4);
  u32* VB  = (u32*)alloc((size_t)N * 4);
  u32* INV = (u32*)alloc((size_t)N * 4);
  u32* P3  = (u32*)alloc((size_t)N * 4);
  u32* P4  = (u32*)alloc((size_t)N * 4);
  u8*  E   = (u8*)alloc((size_t)N);
  u32* histMat = (u32*)alloc((size_t)RADIX * numTiles * 4);
  u32* tileSum = (u32*)alloc((size_t)5 * numTiles * 4);
  u32* digTot  = (u32*)alloc(RADIX * 4);
  u32* digOff  = (u32*)alloc(RADIX * 4);
  u32* dum     = (u32*)alloc(8 * 4);
  // reuse dead sort buffers for three of the five prefix arrays
  u32* P0 = KA; u32* P1 = KB; u32* P2 = VB;

  buildKeysKernel<<<(N + TPB * 4 - 1) / (TPB * 4), TPB, 0, stream>>>(pre, tgt, KA, VA, N);

  u32 *ki = KA, *vi = VA, *ko = KB, *vo = VB;
  for (u32 p = 0; p < 4; p++) {
    histKernel<<<numTiles, TPB, 0, stream>>>(ki, histMat, p * 8, numTiles, N);
    rowScanKernel<<<RADIX, TPB, 0, stream>>>(histMat, numTiles, digTot);
    scanDigitsWMMA<<<1, 32, 0, stream>>>(digTot, digOff);
    if (p < 3)
      scatterKernel<false><<<numTiles, TPB, 0, stream>>>(ki, vi, ko, vo, INV,
                                                         histMat, digOff, p * 8, numTiles, N);
    else
      scatterKernel<true><<<numTiles, TPB, 0, stream>>>(ki, vi, ko, vo, INV,
                                                        histMat, digOff, p * 8, numTiles, N);
    u32* tk = ki; ki = ko; ko = tk;
    u32* tv = vi; vi = vo; vo = tv;
  }
  // after 4 passes: ki==KA (sorted keys, dead), vi==VA holds idx, INV holds inverse

  tileSumKernel<<<numTiles, TPB, 0, stream>>>(INV, level, E, tileSum, numTiles, N);
  rowScanKernel<<<5, TPB, 0, stream>>>(tileSum, numTiles, dum);
  writePKernel<<<numTiles, TPB, 0, stream>>>(E, tileSum, P0, P1, P2, P3, P4, numTiles, N);
  finalKernel<<<(N + TPB - 1) / TPB, TPB, 0, stream>>>(pre, tgt, level, vi,
                                                       P0, P1, P2, P3, P4,
                                                       (float*)d_out, N);
  (void)n_in; (void)out_size; (void)ws_size;
}